// DNANet_7876970020896
// MI455X (gfx1250) — compile-verified
//
#include <hip/hip_runtime.h>
#include <hip/hip_bf16.h>
#include <math.h>

// ---------------------------------------------------------------------------
// Graph-transformer forward for MI455X (gfx1250, wave32).
// Dense node GEMMs: V_WMMA_F32_16X16X4_F32, 32x32 register tile per wave
// (4 WMMA accumulators, A/B fragments reused 2x each -> 2 dwords loaded per
// WMMA instead of 4). Edge attention: one thread per (edge, head), float4
// gathers from L2-resident node tables, hardware fp32 scatter atomics.
// ---------------------------------------------------------------------------

typedef float v2f __attribute__((ext_vector_type(2)));
typedef float v8f __attribute__((ext_vector_type(8)));

#define HID    128
#define OUTC   64
#define HEADS  8
#define DHEAD  16
#define NLAYER 3

// ---------------- small utility kernels ----------------
__global__ void fill_kernel(float* __restrict__ p, float v, int n) {
    int i = blockIdx.x * blockDim.x + threadIdx.x;
    if (i < n) p[i] = v;
}

__global__ void deg_kernel(const int* __restrict__ dst, int E, float* __restrict__ deg) {
    int i = blockIdx.x * blockDim.x + threadIdx.x;
    if (i < E) atomicAdd(&deg[dst[i]], 1.0f);
}

__global__ void rsqrt_kernel(float* __restrict__ d, int n) {
    int i = blockIdx.x * blockDim.x + threadIdx.x;
    if (i < n) d[i] = rsqrtf(d[i]);   // deg >= 1 always (self loop)
}

__global__ void relu_kernel(float* __restrict__ p, int n) {
    int i = blockIdx.x * blockDim.x + threadIdx.x;
    if (i < n) p[i] = fmaxf(p[i], 0.0f);
}

// ---------------- WMMA f32 GEMM:  C[M,Nw] = (relu?)(A[M,K] @ B[K,Nw] + bias)
// One 32x32 output tile per wave = 2x2 grid of 16x16 WMMA tiles.
// A-fragment (ISA 7.12.2, 32-bit A 16x4): lanes 0-15 hold K={k0,k0+1},
// lanes 16-31 hold K={k0+2,k0+3}, M = lane&15. B assumed symmetric
// (N = lane&15, same K split). D: VGPR r -> row r + 8*(lane>>4), col lane&15.
// Requires Nw % 32 == 0 (true: 128 and 64). M clamped/guarded.
__global__ void gemm_wmma_f32(const float* __restrict__ A,
                              const float* __restrict__ B,
                              const float* __restrict__ bias,
                              float* __restrict__ C,
                              int M, int K, int Nw, int relu)
{
    const int lane = threadIdx.x & 31;
    const int wid  = threadIdx.x >> 5;
    const int tilesN2 = Nw >> 5;                 // 32-wide strips
    const int tilesM2 = (M + 31) >> 5;
    const int tile = blockIdx.x * (blockDim.x >> 5) + wid;
    if (tile >= tilesM2 * tilesN2) return;       // whole-wave exit: EXEC all-1s at WMMA

    const int tm = tile / tilesN2;
    const int tn = tile - tm * tilesN2;
    const int hi = lane >> 4;
    const int li = lane & 15;

    const int r0 = (tm << 5) + li;               // A rows this lane loads
    const int r1 = r0 + 16;
    const int r0c = r0 < M ? r0 : (M - 1);       // clamp loads, guard stores
    const int r1c = r1 < M ? r1 : (M - 1);
    const int n0 = (tn << 5) + li;               // B cols this lane loads
    const int n1 = n0 + 16;

    const float* __restrict__ arow0 = A + (size_t)r0c * K;
    const float* __restrict__ arow1 = A + (size_t)r1c * K;

    v8f acc00 = {}, acc01 = {}, acc10 = {}, acc11 = {};
#pragma unroll 4
    for (int k0 = 0; k0 < K; k0 += 4) {
        const int ka = k0 + 2 * hi;
        v2f a0; a0.x = arow0[ka]; a0.y = arow0[ka + 1];
        v2f a1; a1.x = arow1[ka]; a1.y = arow1[ka + 1];
        const float* __restrict__ Bk0 = B + (size_t)ka * Nw;
        const float* __restrict__ Bk1 = Bk0 + Nw;
        v2f b0; b0.x = Bk0[n0]; b0.y = Bk1[n0];
        v2f b1; b1.x = Bk0[n1]; b1.y = Bk1[n1];
        acc00 = __builtin_amdgcn_wmma_f32_16x16x4_f32(false, a0, false, b0, (short)0, acc00, false, false);
        acc01 = __builtin_amdgcn_wmma_f32_16x16x4_f32(false, a0, false, b1, (short)0, acc01, false, false);
        acc10 = __builtin_amdgcn_wmma_f32_16x16x4_f32(false, a1, false, b0, (short)0, acc10, false, false);
        acc11 = __builtin_amdgcn_wmma_f32_16x16x4_f32(false, a1, false, b1, (short)0, acc11, false, false);
    }

    const float bb0 = bias ? bias[n0] : 0.0f;
    const float bb1 = bias ? bias[n1] : 0.0f;
#pragma unroll
    for (int r = 0; r < 8; ++r) {
        const int row0 = (tm << 5) + r + 8 * hi;
        const int row1 = row0 + 16;
        if (row0 < M) {
            float v0 = acc00[r] + bb0;
            float v1 = acc01[r] + bb1;
            if (relu) { v0 = fmaxf(v0, 0.0f); v1 = fmaxf(v1, 0.0f); }
            C[(size_t)row0 * Nw + n0] = v0;
            C[(size_t)row0 * Nw + n1] = v1;
        }
        if (row1 < M) {
            float v0 = acc10[r] + bb0;
            float v1 = acc11[r] + bb1;
            if (relu) { v0 = fmaxf(v0, 0.0f); v1 = fmaxf(v1, 0.0f); }
            C[(size_t)row1 * Nw + n0] = v0;
            C[(size_t)row1 * Nw + n1] = v1;
        }
    }
}

// ---------------- per-(edge, head) attention + scatter --------------------
// Q: [N,128] for dst; Kn/Vn: [Lcur, N, 128] node tables (gathered by src).
// agg must be pre-zeroed; accumulates norm * softmax_l(q.k_l) * v_l.
__global__ void edge_attn_kernel(const int* __restrict__ src,
                                 const int* __restrict__ dst,
                                 int E, int N, int Lcur,
                                 const float* __restrict__ dinv,
                                 const float* __restrict__ Q,
                                 const float* __restrict__ Kn,
                                 const float* __restrict__ Vn,
                                 float* __restrict__ agg)
{
    const long long t = (long long)blockIdx.x * blockDim.x + threadIdx.x;
    const int Et = E + N;
    if (t >= (long long)Et * HEADS) return;
    const int e = (int)(t >> 3);
    const int h = (int)(t & 7);

    int s, d;
    if (e < E) { s = src[e]; d = dst[e]; } else { s = e - E; d = s; }
    const float norm = dinv[s] * dinv[d];

    const float4* __restrict__ qp = (const float4*)(Q + (size_t)d * HID + h * DHEAD);
    const float4 q0 = qp[0], q1 = qp[1], q2 = qp[2], q3 = qp[3];

    float sc[NLAYER];
    float mx = -3.0e38f;
#pragma unroll
    for (int l = 0; l < NLAYER; ++l) {
        if (l >= Lcur) break;
        const float4* __restrict__ kp =
            (const float4*)(Kn + ((size_t)l * N + s) * HID + h * DHEAD);
        const float4 k0 = kp[0], k1 = kp[1], k2 = kp[2], k3 = kp[3];
        float dt = q0.x*k0.x + q0.y*k0.y + q0.z*k0.z + q0.w*k0.w
                 + q1.x*k1.x + q1.y*k1.y + q1.z*k1.z + q1.w*k1.w
                 + q2.x*k2.x + q2.y*k2.y + q2.z*k2.z + q2.w*k2.w
                 + q3.x*k3.x + q3.y*k3.y + q3.z*k3.z + q3.w*k3.w;
        dt *= 0.25f;                       // 1/sqrt(DHEAD)
        sc[l] = dt;
        mx = fmaxf(mx, dt);
    }
    float w[NLAYER];
    float wsum = 0.0f;
#pragma unroll
    for (int l = 0; l < NLAYER; ++l) {
        if (l >= Lcur) break;
        w[l] = expf(sc[l] - mx);
        wsum += w[l];
    }
    const float sfac = norm / wsum;

    float4 m0 = {0,0,0,0}, m1 = {0,0,0,0}, m2 = {0,0,0,0}, m3 = {0,0,0,0};
#pragma unroll
    for (int l = 0; l < NLAYER; ++l) {
        if (l >= Lcur) break;
        const float wl = w[l];
        const float4* __restrict__ vp =
            (const float4*)(Vn + ((size_t)l * N + s) * HID + h * DHEAD);
        const float4 v0 = vp[0], v1 = vp[1], v2 = vp[2], v3 = vp[3];
        m0.x += wl*v0.x; m0.y += wl*v0.y; m0.z += wl*v0.z; m0.w += wl*v0.w;
        m1.x += wl*v1.x; m1.y += wl*v1.y; m1.z += wl*v1.z; m1.w += wl*v1.w;
        m2.x += wl*v2.x; m2.y += wl*v2.y; m2.z += wl*v2.z; m2.w += wl*v2.w;
        m3.x += wl*v3.x; m3.y += wl*v3.y; m3.z += wl*v3.z; m3.w += wl*v3.w;
    }
    // coarse-grained device scratch -> hardware global_atomic_add_f32 is safe
    float* __restrict__ ap = agg + (size_t)d * HID + h * DHEAD;
    unsafeAtomicAdd(ap +  0, m0.x*sfac); unsafeAtomicAdd(ap +  1, m0.y*sfac);
    unsafeAtomicAdd(ap +  2, m0.z*sfac); unsafeAtomicAdd(ap +  3, m0.w*sfac);
    unsafeAtomicAdd(ap +  4, m1.x*sfac); unsafeAtomicAdd(ap +  5, m1.y*sfac);
    unsafeAtomicAdd(ap +  6, m1.z*sfac); unsafeAtomicAdd(ap +  7, m1.w*sfac);
    unsafeAtomicAdd(ap +  8, m2.x*sfac); unsafeAtomicAdd(ap +  9, m2.y*sfac);
    unsafeAtomicAdd(ap + 10, m2.z*sfac); unsafeAtomicAdd(ap + 11, m2.w*sfac);
    unsafeAtomicAdd(ap + 12, m3.x*sfac); unsafeAtomicAdd(ap + 13, m3.y*sfac);
    unsafeAtomicAdd(ap + 14, m3.z*sfac); unsafeAtomicAdd(ap + 15, m3.w*sfac);
}

// ---------------- row-wise log_softmax over 64 logits, one wave32 per row --
__global__ void logsoftmax_kernel(const float* __restrict__ logits,
                                  float* __restrict__ out, int N)
{
    const int gw  = (int)(((long long)blockIdx.x * blockDim.x + threadIdx.x) >> 5);
    const int lane = threadIdx.x & 31;
    if (gw >= N) return;
    const float* __restrict__ row = logits + (size_t)gw * OUTC;
    const float a = row[lane];
    const float b = row[lane + 32];
    float mx = fmaxf(a, b);
#pragma unroll
    for (int off = 16; off > 0; off >>= 1) mx = fmaxf(mx, __shfl_xor(mx, off, 32));
    float sum = expf(a - mx) + expf(b - mx);
#pragma unroll
    for (int off = 16; off > 0; off >>= 1) sum += __shfl_xor(sum, off, 32);
    const float lse = mx + logf(sum);
    out[(size_t)gw * OUTC + lane]      = a - lse;
    out[(size_t)gw * OUTC + lane + 32] = b - lse;
}

// ---------------------------------------------------------------------------
extern "C" void kernel_launch(void* const* d_in, const int* in_sizes, int n_in,
                              void* d_out, int out_size, void* d_ws, size_t ws_size,
                              hipStream_t stream)
{
    const float* x  = (const float*)d_in[0];
    const int*   ei = (const int*)  d_in[1];
    const float* W1 = (const float*)d_in[2];
    const float* b1 = (const float*)d_in[3];
    const float* Wq = (const float*)d_in[4];
    const float* bq = (const float*)d_in[5];
    const float* Wk = (const float*)d_in[6];
    const float* bk = (const float*)d_in[7];
    const float* Wv = (const float*)d_in[8];
    const float* bv = (const float*)d_in[9];
    const float* W2 = (const float*)d_in[10];
    const float* b2 = (const float*)d_in[11];
    float* out = (float*)d_out;

    const int N = in_sizes[0] / HID;       // 10000
    const int E = in_sizes[1] / 2;         // 160000
    const int* srcp = ei;                  // edge_index[0]
    const int* dstp = ei + E;              // edge_index[1]

    // ---- workspace layout (floats), ~59 MB total ----
    float* wsf = (float*)d_ws;
    size_t o = 0;
    float* dinv   = wsf + o; o += (size_t)((N + 255) & ~255);
    float* slots  = wsf + o; o += (size_t)(NLAYER + 1) * N * HID;  // x_all slots 0..3
    float* Qbuf   = wsf + o; o += (size_t)N * HID;
    float* Knbuf  = wsf + o; o += (size_t)NLAYER * N * HID;
    float* Vnbuf  = wsf + o; o += (size_t)NLAYER * N * HID;
    float* logits = wsf + o; o += (size_t)N * OUTC;
    (void)ws_size; (void)n_in; (void)out_size;

    const int TB = 256;                     // 8 wave32 per block
    auto cdiv = [](long long a, long long b) { return (int)((a + b - 1) / b); };

    // degree -> dinv (deg initialized to 1.0 for the self loop)
    fill_kernel <<<cdiv(N, TB), TB, 0, stream>>>(dinv, 1.0f, N);
    deg_kernel  <<<cdiv(E, TB), TB, 0, stream>>>(dstp, E, dinv);
    rsqrt_kernel<<<cdiv(N, TB), TB, 0, stream>>>(dinv, N);

    // 32x32 tiles per wave
    const int tiles128 = ((N + 31) / 32) * (HID / 32);   // Nw = 128
    const int tiles64  = ((N + 31) / 32) * (OUTC / 32);  // Nw = 64

    // h = relu(x @ W1 + b1) -> slot 0
    gemm_wmma_f32<<<cdiv(tiles128, 8), TB, 0, stream>>>(
        x, W1, b1, slots, N, HID, HID, /*relu=*/1);

    for (int l = 0; l < NLAYER; ++l) {
        const int Lcur = l + 1;
        const float* xlast = slots + (size_t)l * N * HID;

        // Q = x_last @ Wq[l] + bq[l]
        gemm_wmma_f32<<<cdiv(tiles128, 8), TB, 0, stream>>>(
            xlast, Wq + (size_t)l * HID * HID, bq + (size_t)l * HID,
            Qbuf, N, HID, HID, 0);

        // K,V node tables for every slot
        for (int s = 0; s < Lcur; ++s) {
            const float* xs = slots + (size_t)s * N * HID;
            gemm_wmma_f32<<<cdiv(tiles128, 8), TB, 0, stream>>>(
                xs, Wk + (size_t)l * HID * HID, bk + (size_t)l * HID,
                Knbuf + (size_t)s * N * HID, N, HID, HID, 0);
            gemm_wmma_f32<<<cdiv(tiles128, 8), TB, 0, stream>>>(
                xs, Wv + (size_t)l * HID * HID, bv + (size_t)l * HID,
                Vnbuf + (size_t)s * N * HID, N, HID, HID, 0);
        }

        // aggregate into next slot
        float* nxt = slots + (size_t)(l + 1) * N * HID;
        fill_kernel<<<cdiv((long long)N * HID, TB), TB, 0, stream>>>(nxt, 0.0f, N * HID);
        const long long tasks = (long long)(E + N) * HEADS;
        edge_attn_kernel<<<cdiv(tasks, TB), TB, 0, stream>>>(
            srcp, dstp, E, N, Lcur, dinv, Qbuf, Knbuf, Vnbuf, nxt);
        relu_kernel<<<cdiv((long long)N * HID, TB), TB, 0, stream>>>(nxt, N * HID);
    }

    // logits = slot3 @ W2 + b2 ; out = log_softmax(logits)
    const float* xfin = slots + (size_t)NLAYER * N * HID;
    gemm_wmma_f32<<<cdiv(tiles64, 8), TB, 0, stream>>>(
        xfin, W2, b2, logits, N, HID, OUTC, 0);
    logsoftmax_kernel<<<cdiv((long long)N * 32, TB), TB, 0, stream>>>(logits, out, N);
}